// Autoencoder_36464272343198
// MI455X (gfx1250) — compile-verified
//
#include <hip/hip_runtime.h>
#include <hip/hip_bf16.h>
#include <math.h>

// MI455X / gfx1250, wave32. GEMMs use V_WMMA_F32_16X16X4_F32 (full fp32
// precision through the matrix pipe; the scatter phase dominates runtime so
// fp32 WMMA is within ~2x of the HBM roofline and carries zero accuracy risk).

typedef __attribute__((ext_vector_type(2))) float v2f;
typedef __attribute__((ext_vector_type(8))) float v8f;

#define IN_SIZE 512
#define HID1    128

// ---------------------------------------------------------------------------
// acc = 0, deg = 1 (self-loop contributes 1 to every node's degree).
// Must run every call: the harness does not re-zero ws between graph replays.
__global__ void init_ws_kernel(float* __restrict__ acc, float* __restrict__ deg,
                               int accN, int nodes) {
  int i = blockIdx.x * blockDim.x + threadIdx.x;
  if (i < accN)  acc[i] = 0.0f;
  if (i < nodes) deg[i] = 1.0f;
}

// deg[col[e]] += 1 over all edges (in-degree + self-loop).
__global__ void degree_kernel(const int* __restrict__ col,
                              float* __restrict__ deg, int E) {
  int e = blockIdx.x * blockDim.x + threadIdx.x;
  if (e < E) atomicAdd(&deg[col[e]], 1.0f);
}

// In-place deg -> rsqrt(deg). deg >= 1 always, so no zero-guard needed.
__global__ void rsqrt_kernel(float* __restrict__ deg, int n) {
  int i = blockIdx.x * blockDim.x + threadIdx.x;
  if (i < n) deg[i] = rsqrtf(deg[i]);
}

// Edge scatter: acc[col] += dinv[row]*dinv[col] * h[row].
// 32 threads per edge, 4 features each -> 128B coalesced gathers and
// coalesced global_atomic_add_f32 bursts; h/acc (25.6 MB each) live in L2.
__global__ __launch_bounds__(256) void scatter_kernel(
    const int* __restrict__ row, const int* __restrict__ col,
    const float* __restrict__ dinv, const float* __restrict__ h,
    float* __restrict__ acc, int E) {
  unsigned int gid = blockIdx.x * 256u + threadIdx.x;
  unsigned int e = gid >> 5;
  if (e >= (unsigned int)E) return;
  int f = (int)(gid & 31u) * 4;
  int r = row[e], c = col[e];
  float norm = dinv[r] * dinv[c];
  const float4 hv = *(const float4*)(h + (size_t)r * HID1 + f);
  float* dst = acc + (size_t)c * HID1 + f;
  atomicAdd(dst + 0, norm * hv.x);
  atomicAdd(dst + 1, norm * hv.y);
  atomicAdd(dst + 2, norm * hv.z);
  atomicAdd(dst + 3, norm * hv.w);
}

// ---------------------------------------------------------------------------
// Tiled GEMM: C[M,Ntot] = act(Aeff[M,K] @ W[K,Ntot] + bias).
// 16 waves/block (512 threads): block tile = 256 rows x 128 cols, wave tile =
// 16 rows x 128 cols. 16 waves over 4 SIMD32s = 4 waves/SIMD for latency
// hiding even when the 288KB encoder panel limits us to 1 block/WGP; also
// halves the number of blocks -> halves redundant W-panel traffic from L2.
//
// WMMA fp32 16x16x4 lane layouts (wave32, ISA 7.12.2):
//   A (16x4): lane L holds A[M=L%16][k0 + 2*(L>>4) + {0,1}] in V0,V1.
//   B (4x16): lane L holds B[k0 + 2*(L>>4) + {0,1}][N=L%16] in V0,V1.
//   C (16x16): VGPR r: lanes 0-15 -> M=r, lanes 16-31 -> M=r+8; N = lane%16.
//
// W panel in LDS as K/2 "pair rows" of float2 {W[2p][n], W[2p+1][n]}, padded
// to 144 float2 per pair row (1152 B; 1152 mod 256 == 128) so the two lane
// halves (pair rows p and p+1) hit disjoint 32-bank windows -> conflict-free
// ds_load_b64 (compiler fuses them into ds_load_2addr_b64).
//
// FUSE (decoder): Aeff = acc + dinv[row]^2 * h + gcn_bias[k]; dinv^2 hoisted
// since each lane's row is fixed across the K loop.
template <int K, int ACT, bool FUSE>
__global__ __launch_bounds__(512) void gemm_wmma_kernel(
    const float* __restrict__ A,     // [M,K]
    const float* __restrict__ H,     // [M,K]   (FUSE only)
    const float* __restrict__ dinv,  // [M]     (FUSE only)
    const float* __restrict__ gb,    // [K]     (FUSE only, gcn bias)
    const float* __restrict__ W,     // [K,Ntot]
    const float* __restrict__ bias,  // [Ntot]
    float* __restrict__ C,           // [M,Ntot]
    int M, int Ntot) {
  constexpr int PAIRS = K / 2;
  constexpr int PSTR2 = 144 * 2;               // floats per pair row (padded)
  __shared__ float lds[PAIRS * PSTR2];         // K=512: 288KB; K=128: 72KB

  const int tid = threadIdx.x;
  const int nChunk = blockIdx.y * 128;

  // Cooperative W panel fill (lane n writes stride-2 banks of parity k&1 -> no conflicts).
  for (int idx = tid; idx < K * 128; idx += 512) {
    int k = idx >> 7, n = idx & 127;
    lds[(k >> 1) * PSTR2 + n * 2 + (k & 1)] = W[(size_t)k * Ntot + nChunk + n];
  }
  __syncthreads();

  const int wave = tid >> 5, lane = tid & 31;
  const int lrow = lane & 15, half = lane >> 4;
  const int mbase = blockIdx.x * 256 + wave * 16;
  int row = mbase + lrow;
  if (row > M - 1) row = M - 1;                // clamp loads; stores guarded
  const float* Arow = A + (size_t)row * K;
  const float* Hrow = FUSE ? (H + (size_t)row * K) : nullptr;
  float selfw = 0.0f;
  if (FUSE) { float d = dinv[row]; selfw = d * d; }

  v8f accv[8];
  const v8f vzero = {0.f, 0.f, 0.f, 0.f, 0.f, 0.f, 0.f, 0.f};
#pragma unroll
  for (int t = 0; t < 8; ++t) accv[t] = vzero;

  for (int k0 = 0; k0 < K; k0 += 4) {
    const int ka = k0 + half * 2;
    v2f a;
    a.x = Arow[ka];
    a.y = Arow[ka + 1];
    if (FUSE) {
      a.x += selfw * Hrow[ka]     + gb[ka];
      a.y += selfw * Hrow[ka + 1] + gb[ka + 1];
    }
    const int prow = (k0 >> 1) + half;
    const float* bbase = lds + prow * PSTR2 + lrow * 2;
#pragma unroll
    for (int t = 0; t < 8; ++t) {
      v2f b = *(const v2f*)(bbase + t * 32);   // (t*16 + lrow) float2
      accv[t] = __builtin_amdgcn_wmma_f32_16x16x4_f32(
          false, a, false, b, (short)0, accv[t], false, false);
    }
  }

  // Epilogue: bias + activation; 64B-contiguous stores per half-wave.
#pragma unroll
  for (int t = 0; t < 8; ++t) {
    const int colIdx = nChunk + t * 16 + lrow;
    const float bv = bias[colIdx];
#pragma unroll
    for (int r = 0; r < 8; ++r) {
      const int orow = mbase + r + half * 8;
      if (orow < M) {
        float v = accv[t][r] + bv;
        if (ACT == 0) v = fmaxf(v, 0.0f);                 // relu
        else          v = 1.0f / (1.0f + __expf(-v));     // sigmoid
        C[(size_t)orow * Ntot + colIdx] = v;
      }
    }
  }
}

// ---------------------------------------------------------------------------
extern "C" void kernel_launch(void* const* d_in, const int* in_sizes, int n_in,
                              void* d_out, int out_size, void* d_ws, size_t ws_size,
                              hipStream_t stream) {
  const float* x     = (const float*)d_in[0];
  const float* W_enc = (const float*)d_in[1];
  const float* b_enc = (const float*)d_in[2];
  const float* W_dec = (const float*)d_in[3];
  const float* b_dec = (const float*)d_in[4];
  const float* gcnb  = (const float*)d_in[5];
  const int*   eidx  = (const int*)d_in[6];
  float* out = (float*)d_out;

  const int N = in_sizes[0] / IN_SIZE;  // 50000 nodes
  const int E = in_sizes[6] / 2;        // 640000 edges
  const int* erow = eidx;               // edge_index[0] = source nodes
  const int* ecol = eidx + E;           // edge_index[1] = aggregation targets

  // Workspace layout: h [N,128] | acc [N,128] | deg/dinv [N]  (~51.6 MB)
  float* h    = (float*)d_ws;
  float* acc  = h + (size_t)N * HID1;
  float* dinv = acc + (size_t)N * HID1;

  const int accN = N * HID1;
  init_ws_kernel<<<(accN + 255) / 256, 256, 0, stream>>>(acc, dinv, accN, N);
  degree_kernel<<<(E + 255) / 256, 256, 0, stream>>>(ecol, dinv, E);
  rsqrt_kernel<<<(N + 255) / 256, 256, 0, stream>>>(dinv, N);

  const int mBlocks = (N + 255) / 256;

  // h = relu(x @ W_enc + b_enc): M=50000, K=512, N=128 (one N-chunk).
  gemm_wmma_kernel<IN_SIZE, 0, false><<<dim3(mBlocks, 1), 512, 0, stream>>>(
      x, nullptr, nullptr, nullptr, W_enc, b_enc, h, N, HID1);

  // acc[col] += dinv[row]*dinv[col] * h[row] over all edges.
  scatter_kernel<<<(E * 32 + 255) / 256, 256, 0, stream>>>(
      erow, ecol, dinv, h, acc, E);

  // out = sigmoid((acc + dinv^2*h + gcn_bias) @ W_dec + b_dec):
  // M=50000, K=128, N=512 (4 N-chunks).
  gemm_wmma_kernel<HID1, 1, true><<<dim3(mBlocks, IN_SIZE / 128), 512, 0, stream>>>(
      acc, h, dinv, gcnb, W_dec, b_dec, out, N, IN_SIZE);
}